// TwoDir2LayerGraphConvolution_50113678409787
// MI455X (gfx1250) — compile-verified
//
#include <hip/hip_runtime.h>

typedef __attribute__((ext_vector_type(2))) float v2f;
typedef __attribute__((ext_vector_type(8))) float v8f;

#define NODES 100000
#define FDIM  128
#define HDIM  128

// ---------------------------------------------------------------------------
// GEMM1: out[N,128] = A[N,K] @ W[K,128] + bias   (fp32 WMMA 16x16x4)
// block = 256 threads = 8 waves; each wave owns one 16x16 output tile.
// blockIdx.x -> 16-row node tile (N = 100000 = 6250*16 exactly).
// ---------------------------------------------------------------------------
__global__ __launch_bounds__(256)
void gcn_gemm_f32(const float* __restrict__ A, const float* __restrict__ W,
                  const float* __restrict__ bias, float* __restrict__ out,
                  int K) {
    const int wave = threadIdx.x >> 5;      // 0..7 : column tile
    const int lane = threadIdx.x & 31;
    const int half = lane >> 4;             // 0/1
    const int mn   = lane & 15;             // m for A-frag, n for B/C-frag
    const int row0 = blockIdx.x * 16;
    const int col0 = wave * 16;

    v8f c = {};
    const float* Arow = A + (size_t)(row0 + mn) * K;
    for (int kk = 0; kk < K; kk += 4) {
        const int ka = kk + 2 * half;       // this lane covers K = ka, ka+1
        v2f a; a.x = Arow[ka];
               a.y = Arow[ka + 1];
        v2f b; b.x = W[(size_t)ka       * HDIM + col0 + mn];
               b.y = W[(size_t)(ka + 1) * HDIM + col0 + mn];
        c = __builtin_amdgcn_wmma_f32_16x16x4_f32(false, a, false, b,
                                                  (short)0, c, false, false);
    }
    const float bv = bias[col0 + mn];
    float* orow = out + (size_t)row0 * HDIM + col0 + mn;
#pragma unroll
    for (int j = 0; j < 8; ++j)
        orow[(size_t)(j + 8 * half) * HDIM] = c[j] + bv;
}

// ---------------------------------------------------------------------------
// GEMM2: out[N,128] = relu(concat(X0,X1,X2))[N,384] @ W2[384,128] + b2
// Same WMMA structure, K=384, relu+concat fused into the A-fragment load.
// ---------------------------------------------------------------------------
__global__ __launch_bounds__(256)
void gcn_gemm2_concat_relu(const float* __restrict__ X0,
                           const float* __restrict__ X1,
                           const float* __restrict__ X2,
                           const float* __restrict__ W2,
                           const float* __restrict__ bias,
                           float* __restrict__ out) {
    const int wave = threadIdx.x >> 5;
    const int lane = threadIdx.x & 31;
    const int half = lane >> 4;
    const int mn   = lane & 15;
    const int row0 = blockIdx.x * 16;
    const int col0 = wave * 16;

    v8f c = {};
    for (int kk = 0; kk < 3 * HDIM; kk += 4) {
        const int ka = kk + 2 * half;                 // ka, ka+1 same segment
        const float* Xs = (ka < 128) ? X0 : (ka < 256) ? X1 : X2;
        const float* r  = Xs + (size_t)(row0 + mn) * HDIM + (ka & 127);
        v2f a; a.x = fmaxf(r[0], 0.0f);
               a.y = fmaxf(r[1], 0.0f);
        v2f b; b.x = W2[(size_t)ka       * HDIM + col0 + mn];
               b.y = W2[(size_t)(ka + 1) * HDIM + col0 + mn];
        c = __builtin_amdgcn_wmma_f32_16x16x4_f32(false, a, false, b,
                                                  (short)0, c, false, false);
    }
    const float bv = bias[col0 + mn];
    float* orow = out + (size_t)row0 * HDIM + col0 + mn;
#pragma unroll
    for (int j = 0; j < 8; ++j)
        orow[(size_t)(j + 8 * half) * HDIM] = c[j] + bv;
}

// ---------------------------------------------------------------------------
// SpMM scatter: out[r,:] += v * x[c,:]  per edge. One wave per edge:
// 32 lanes * float4 = 128 floats; coalesced gather, f32 atomics to L2.
// ---------------------------------------------------------------------------
__global__ __launch_bounds__(256)
void gcn_spmm_atomic(const int* __restrict__ rows, const int* __restrict__ cols,
                     const float* __restrict__ vals, const float* __restrict__ x,
                     float* __restrict__ out, int E) {
    const int lane = threadIdx.x & 31;
    const int wid  = (int)((blockIdx.x * blockDim.x + threadIdx.x) >> 5);
    const int nw   = (int)((gridDim.x * blockDim.x) >> 5);
    for (int e = wid; e < E; e += nw) {
        const int   r = rows[e];
        const int   cc = cols[e];
        const float v = vals[e];
        const float4 xv = ((const float4*)(x + (size_t)cc * HDIM))[lane];
        float* o = out + (size_t)r * HDIM + lane * 4;
        atomicAdd(o + 0, v * xv.x);
        atomicAdd(o + 1, v * xv.y);
        atomicAdd(o + 2, v * xv.z);
        atomicAdd(o + 3, v * xv.w);
    }
}

// ---------------------------------------------------------------------------
// Head: out[q,:] = concat(relu(x2[e0]), relu(x2[e1]))[768] @ Wl[768,2] + bl
// One wave per query, wave32 shuffle reduction.
// ---------------------------------------------------------------------------
__global__ __launch_bounds__(256)
void gcn_head(const float* __restrict__ X0, const float* __restrict__ X1,
              const float* __restrict__ X2, const int* __restrict__ qe,
              const float* __restrict__ Wl, const float* __restrict__ bl,
              float* __restrict__ out, int Q) {
    const int lane = threadIdx.x & 31;
    const int q    = (int)((blockIdx.x * blockDim.x + threadIdx.x) >> 5);
    if (q >= Q) return;
    const int n0 = qe[q * 2 + 0];
    const int n1 = qe[q * 2 + 1];
    float acc0 = 0.0f, acc1 = 0.0f;
    const float* bufs[3] = {X0, X1, X2};
#pragma unroll
    for (int s = 0; s < 6; ++s) {
        const int    node = (s < 3) ? n0 : n1;
        const float* X    = bufs[s % 3];
        const float4 xv   = ((const float4*)(X + (size_t)node * HDIM))[lane];
        const int kb = s * HDIM + lane * 4;
        const float a0 = fmaxf(xv.x, 0.0f), a1 = fmaxf(xv.y, 0.0f);
        const float a2 = fmaxf(xv.z, 0.0f), a3 = fmaxf(xv.w, 0.0f);
        acc0 += a0 * Wl[(kb + 0) * 2 + 0] + a1 * Wl[(kb + 1) * 2 + 0]
              + a2 * Wl[(kb + 2) * 2 + 0] + a3 * Wl[(kb + 3) * 2 + 0];
        acc1 += a0 * Wl[(kb + 0) * 2 + 1] + a1 * Wl[(kb + 1) * 2 + 1]
              + a2 * Wl[(kb + 2) * 2 + 1] + a3 * Wl[(kb + 3) * 2 + 1];
    }
#pragma unroll
    for (int off = 16; off > 0; off >>= 1) {
        acc0 += __shfl_down(acc0, off, 32);
        acc1 += __shfl_down(acc1, off, 32);
    }
    if (lane == 0) {
        out[q * 2 + 0] = acc0 + bl[0];
        out[q * 2 + 1] = acc1 + bl[1];
    }
}

extern "C" void kernel_launch(void* const* d_in, const int* in_sizes, int n_in,
                              void* d_out, int out_size, void* d_ws, size_t ws_size,
                              hipStream_t stream) {
    const float* un_f  = (const float*)d_in[0];
    const float* in_f  = (const float*)d_in[1];
    const float* out_f = (const float*)d_in[2];
    const int*   rws[3] = {(const int*)d_in[3], (const int*)d_in[6], (const int*)d_in[9]};
    const int*   cls[3] = {(const int*)d_in[4], (const int*)d_in[7], (const int*)d_in[10]};
    const float* vls[3] = {(const float*)d_in[5], (const float*)d_in[8], (const float*)d_in[11]};
    const int*   qedges = (const int*)d_in[12];
    const float* W1 = (const float*)d_in[13];
    const float* b1 = (const float*)d_in[14];
    const float* W2 = (const float*)d_in[15];
    const float* b2 = (const float*)d_in[16];
    const float* Wl = (const float*)d_in[17];
    const float* bl = (const float*)d_in[18];
    float* out = (float*)d_out;

    const int N = in_sizes[0] / FDIM;      // 100000
    const int E = in_sizes[3];             // 1600000
    const int Q = in_sizes[12] / 2;        // 100000

    const size_t NB = (size_t)N * HDIM;    // floats per node matrix
    float* ws = (float*)d_ws;
    float* s0 = ws + 0 * NB;   // y_un, later h
    float* s1 = ws + 1 * NB;   // y_in, later x2_un
    float* s2 = ws + 2 * NB;   // y_out, later x2_in
    float* s3 = ws + 3 * NB;   // x1_un, later x2_out
    float* s4 = ws + 4 * NB;   // x1_in
    float* s5 = ws + 5 * NB;   // x1_out
    float* yb[3]  = {s0, s1, s2};
    float* x1b[3] = {s3, s4, s5};
    const float* feats[3] = {un_f, in_f, out_f};

    const int mtiles = N / 16;             // 6250
    const int spmm_blocks = 2048;          // 16384 waves over 1.6M edges

    // layer 1 projections: y_dir = feat_dir @ W1 + b1
    for (int d = 0; d < 3; ++d)
        gcn_gemm_f32<<<mtiles, 256, 0, stream>>>(feats[d], W1, b1, yb[d], FDIM);

    // spmm pass 1 (zero accumulators, then scatter-add)
    hipMemsetAsync(s3, 0, 3 * NB * sizeof(float), stream);
    for (int d = 0; d < 3; ++d)
        gcn_spmm_atomic<<<spmm_blocks, 256, 0, stream>>>(rws[d], cls[d], vls[d],
                                                         yb[d], x1b[d], E);

    // h = relu(concat(x1)) @ W2 + b2  -> s0
    gcn_gemm2_concat_relu<<<mtiles, 256, 0, stream>>>(s3, s4, s5, W2, b2, s0);

    // spmm pass 2: propagate shared h through all three adjacencies
    float* x2b[3] = {s1, s2, s3};
    hipMemsetAsync(s1, 0, 3 * NB * sizeof(float), stream);
    for (int d = 0; d < 3; ++d)
        gcn_spmm_atomic<<<spmm_blocks, 256, 0, stream>>>(rws[d], cls[d], vls[d],
                                                         s0, x2b[d], E);

    // edge head: gather endpoints, relu, [Q,768] @ [768,2] + bl
    const int head_blocks = (Q * 32 + 255) / 256;
    gcn_head<<<head_blocks, 256, 0, stream>>>(s1, s2, s3, qedges, Wl, bl, out, Q);
}